// TGCNActor_3985729651217
// MI455X (gfx1250) — compile-verified
//
#include <hip/hip_runtime.h>
#include <hip/hip_bf16.h>

// ---------------------------------------------------------------------------
// T-GCN actor on gfx1250 (MI455X).
//  - fp32 WMMA (V_WMMA_F32_16X16X4_F32) for all big GEMMs (reference is fp32,
//    argmax output is sensitive to precision).
//  - Double-buffered LDS staging via GLOBAL_LOAD_ASYNC_TO_LDS_B128 (ASYNCcnt),
//    overlapping global->LDS DMA with WMMA compute.
// ---------------------------------------------------------------------------

typedef float v2f __attribute__((ext_vector_type(2)));
typedef float v8f __attribute__((ext_vector_type(8)));

#define Nn 1024
#define Tt 32
#define Dd 256
#define Hh 1024
#define Ee 64

#define BM 128
#define BN 128
#define BK 32
#define PADA 4
#define PADB 4
#define LDA_S (BK + PADA)   // 36 floats -> 144B rows (16B aligned)
#define LDB_S (BN + PADB)   // 132 floats -> 528B rows (16B aligned)

static __device__ __forceinline__ float sigmoidf_(float v) {
    return 1.0f / (1.0f + __expf(-v));
}

// Async 16-byte global -> LDS copy (per lane), tracked by ASYNCcnt.
// For a generic pointer into __shared__, addr[31:0] IS the LDS offset
// (flat aperture rule), so truncation gives the VDST LDS address.
static __device__ __forceinline__ void async_b128(const float* g, float* lds_p) {
    unsigned loff = (unsigned)(uintptr_t)lds_p;
    asm volatile("global_load_async_to_lds_b128 %0, %1, off"
                 :: "v"(loff), "v"(g) : "memory");
}
static __device__ __forceinline__ void wait_async0() {
    asm volatile("s_wait_asynccnt 0" ::: "memory");
}

// ---------------------------------------------------------------------------
// GEMM-1: P = x_t @ W[0:256,:] + hsrc @ W[256:1280,:]      (1024 x 1024)
//   x: (N, T, D) row-major, hsrc: (N, H), W: (D+H, H), P: (N, H)
// ---------------------------------------------------------------------------
__global__ __launch_bounds__(256) void gemm_concat_kernel(
    const float* __restrict__ x, int t, const float* __restrict__ hsrc,
    const float* __restrict__ W, float* __restrict__ P)
{
    __shared__ __align__(16) float sA[2][BM * LDA_S];
    __shared__ __align__(16) float sB[2][BK * LDB_S];

    const int tid  = threadIdx.x;
    const int lane = tid & 31;
    const int wid  = tid >> 5;          // 0..7 (wave32)
    const int wm   = (wid & 1) * 64;    // wave M offset in block tile
    const int wn   = (wid >> 1) * 32;   // wave N offset in block tile
    const int m0   = blockIdx.y * BM;
    const int n0   = blockIdx.x * BN;
    const int l16  = lane & 15;
    const int half = lane >> 4;

    v8f acc[4][2];
    #pragma unroll
    for (int i = 0; i < 4; ++i)
        #pragma unroll
        for (int j = 0; j < 2; ++j)
            #pragma unroll
            for (int e = 0; e < 8; ++e) acc[i][j][e] = 0.0f;

    auto stage = [&](int k0, int buf) {
        #pragma unroll
        for (int i = 0; i < 4; ++i) {           // A tile: BM x BK
            int lin = tid + i * 256;            // 1024 float4 slots
            int row = lin >> 3;                 // 8 float4 per row
            int c4  = (lin & 7) * 4;
            int gm  = m0 + row;
            int gk  = k0 + c4;
            const float* g = (gk < Dd)
                ? (x + ((size_t)gm * Tt + t) * Dd + gk)
                : (hsrc + (size_t)gm * Hh + (gk - Dd));
            async_b128(g, &sA[buf][row * LDA_S + c4]);
        }
        #pragma unroll
        for (int i = 0; i < 4; ++i) {           // B tile: BK x BN
            int lin = tid + i * 256;
            int row = lin >> 5;                 // 32 float4 per row
            int c4  = (lin & 31) * 4;
            async_b128(W + (size_t)(k0 + row) * Hh + n0 + c4,
                       &sB[buf][row * LDB_S + c4]);
        }
    };

    const int KTOT = Dd + Hh;           // 1280, multiple of BK
    stage(0, 0);
    wait_async0();
    __syncthreads();
    int p = 0;

    for (int k0 = 0; k0 < KTOT; k0 += BK) {
        if (k0 + BK < KTOT) stage(k0 + BK, p ^ 1);   // prefetch next chunk

        #pragma unroll
        for (int ks = 0; ks < BK / 4; ++ks) {
            const int kk = ks * 4 + half * 2;   // fp32 A/B VGPR layout
            v2f a[4];
            #pragma unroll
            for (int mt = 0; mt < 4; ++mt)
                a[mt] = *(const v2f*)(&sA[p][(wm + mt * 16 + l16) * LDA_S + kk]);
            v2f b[2];
            #pragma unroll
            for (int nt = 0; nt < 2; ++nt) {
                const float* q = &sB[p][kk * LDB_S + wn + nt * 16 + l16];
                b[nt].x = q[0];
                b[nt].y = q[LDB_S];
            }
            #pragma unroll
            for (int mt = 0; mt < 4; ++mt)
                #pragma unroll
                for (int nt = 0; nt < 2; ++nt)
                    acc[mt][nt] = __builtin_amdgcn_wmma_f32_16x16x4_f32(
                        false, a[mt], false, b[nt], (short)0, acc[mt][nt],
                        false, false);
        }
        wait_async0();      // prefetch landed
        __syncthreads();    // all waves done reading buffer p
        p ^= 1;
    }

    #pragma unroll
    for (int mt = 0; mt < 4; ++mt)
        #pragma unroll
        for (int nt = 0; nt < 2; ++nt) {
            int col = n0 + wn + nt * 16 + l16;
            #pragma unroll
            for (int v = 0; v < 8; ++v) {
                int row = m0 + wm + mt * 16 + v + half * 8;
                P[(size_t)row * Hh + col] = acc[mt][nt][v];
            }
        }
}

// ---------------------------------------------------------------------------
// GEMM-2: C = A @ adj + bias, fused epilogues:
//   mode 0: dst = sigmoid(C)                 (z gate)
//   mode 1: dst = sigmoid(C) * hbuf          (r gate * h, materialized as rh)
//   mode 2: hbuf = sigmoid(z*h + (1-z)*tanh(C))   (in-place h update)
// ---------------------------------------------------------------------------
__global__ __launch_bounds__(256) void gemm_adj_kernel(
    const float* __restrict__ A, const float* __restrict__ adj,
    const float* __restrict__ bias, const float* __restrict__ zbuf,
    float* hbuf, float* dst, int mode)
{
    __shared__ __align__(16) float sA[2][BM * LDA_S];
    __shared__ __align__(16) float sB[2][BK * LDB_S];

    const int tid  = threadIdx.x;
    const int lane = tid & 31;
    const int wid  = tid >> 5;
    const int wm   = (wid & 1) * 64;
    const int wn   = (wid >> 1) * 32;
    const int m0   = blockIdx.y * BM;
    const int n0   = blockIdx.x * BN;
    const int l16  = lane & 15;
    const int half = lane >> 4;

    v8f acc[4][2];
    #pragma unroll
    for (int i = 0; i < 4; ++i)
        #pragma unroll
        for (int j = 0; j < 2; ++j)
            #pragma unroll
            for (int e = 0; e < 8; ++e) acc[i][j][e] = 0.0f;

    auto stage = [&](int k0, int buf) {
        #pragma unroll
        for (int i = 0; i < 4; ++i) {
            int lin = tid + i * 256;
            int row = lin >> 3;
            int c4  = (lin & 7) * 4;
            async_b128(A + (size_t)(m0 + row) * Hh + k0 + c4,
                       &sA[buf][row * LDA_S + c4]);
        }
        #pragma unroll
        for (int i = 0; i < 4; ++i) {
            int lin = tid + i * 256;
            int row = lin >> 5;
            int c4  = (lin & 31) * 4;
            async_b128(adj + (size_t)(k0 + row) * Nn + n0 + c4,
                       &sB[buf][row * LDB_S + c4]);
        }
    };

    stage(0, 0);
    wait_async0();
    __syncthreads();
    int p = 0;

    for (int k0 = 0; k0 < Hh; k0 += BK) {
        if (k0 + BK < Hh) stage(k0 + BK, p ^ 1);

        #pragma unroll
        for (int ks = 0; ks < BK / 4; ++ks) {
            const int kk = ks * 4 + half * 2;
            v2f a[4];
            #pragma unroll
            for (int mt = 0; mt < 4; ++mt)
                a[mt] = *(const v2f*)(&sA[p][(wm + mt * 16 + l16) * LDA_S + kk]);
            v2f b[2];
            #pragma unroll
            for (int nt = 0; nt < 2; ++nt) {
                const float* q = &sB[p][kk * LDB_S + wn + nt * 16 + l16];
                b[nt].x = q[0];
                b[nt].y = q[LDB_S];
            }
            #pragma unroll
            for (int mt = 0; mt < 4; ++mt)
                #pragma unroll
                for (int nt = 0; nt < 2; ++nt)
                    acc[mt][nt] = __builtin_amdgcn_wmma_f32_16x16x4_f32(
                        false, a[mt], false, b[nt], (short)0, acc[mt][nt],
                        false, false);
        }
        wait_async0();
        __syncthreads();
        p ^= 1;
    }

    #pragma unroll
    for (int mt = 0; mt < 4; ++mt)
        #pragma unroll
        for (int nt = 0; nt < 2; ++nt) {
            int col = n0 + wn + nt * 16 + l16;
            float bcol = bias[col];
            #pragma unroll
            for (int v = 0; v < 8; ++v) {
                int row = m0 + wm + mt * 16 + v + half * 8;
                size_t idx = (size_t)row * Hh + col;
                float val = acc[mt][nt][v] + bcol;
                if (mode == 0) {
                    dst[idx] = sigmoidf_(val);
                } else if (mode == 1) {
                    dst[idx] = sigmoidf_(val) * hbuf[idx];
                } else {
                    float ht = tanhf(val);
                    float z  = zbuf[idx];
                    float hv = hbuf[idx];
                    hbuf[idx] = sigmoidf_(z * hv + (1.0f - z) * ht);
                }
            }
        }
}

// ---------------------------------------------------------------------------
// Actor head: logits = h @ fc_w + fc_b; out[n*T + t] = argmax_j(logits[n][j])
// (softmax is monotonic -> argmax over logits; ties keep first index)
// One 64-thread block (2 waves) per node.
// ---------------------------------------------------------------------------
__global__ __launch_bounds__(64) void actor_kernel(
    const float* __restrict__ h, const float* __restrict__ fcw,
    const float* __restrict__ fcb, float* __restrict__ out, int t)
{
    __shared__ float sval[Ee];
    __shared__ int   sidx[Ee];

    const int n = blockIdx.x;
    const int j = threadIdx.x;

    float accv = fcb[j];
    const float* hr = h + (size_t)n * Hh;
    for (int k = 0; k < Hh; k += 4) {
        float4 hv = *(const float4*)(hr + k);
        accv += hv.x * fcw[(size_t)(k + 0) * Ee + j];
        accv += hv.y * fcw[(size_t)(k + 1) * Ee + j];
        accv += hv.z * fcw[(size_t)(k + 2) * Ee + j];
        accv += hv.w * fcw[(size_t)(k + 3) * Ee + j];
    }
    sval[j] = accv;
    sidx[j] = j;
    __syncthreads();
    for (int s = 32; s > 0; s >>= 1) {
        if (j < s) {
            if (sval[j + s] > sval[j]) {  // strict > keeps first-max index
                sval[j] = sval[j + s];
                sidx[j] = sidx[j + s];
            }
        }
        __syncthreads();
    }
    if (j == 0) out[(size_t)n * Tt + t] = (float)sidx[0];
}

// ---------------------------------------------------------------------------
extern "C" void kernel_launch(void* const* d_in, const int* in_sizes, int n_in,
                              void* d_out, int out_size, void* d_ws, size_t ws_size,
                              hipStream_t stream) {
    const float* x   = (const float*)d_in[0];   // (N, T, D)
    const float* adj = (const float*)d_in[1];   // (N, N)
    const float* Wz  = (const float*)d_in[2];   // (D+H, H)
    const float* bz  = (const float*)d_in[3];
    const float* Wr  = (const float*)d_in[4];
    const float* br  = (const float*)d_in[5];
    const float* Wh  = (const float*)d_in[6];
    const float* bh  = (const float*)d_in[7];
    const float* fcw = (const float*)d_in[8];   // (H, E)
    const float* fcb = (const float*)d_in[9];
    float* out = (float*)d_out;                 // (N, T, 1) action indices
    float* ws  = (float*)d_ws;

    const size_t mat = (size_t)Nn * Hh;         // 1M floats
    float* hbuf = ws;                           // h state, in-place updated
    float* Pz   = ws + 1 * mat;
    float* Pr   = ws + 2 * mat;
    float* Zb   = ws + 3 * mat;
    float* RH   = ws + 4 * mat;                 // r * h
    float* Ph   = ws + 5 * mat;                 // total 24 MB of d_ws

    hipMemsetAsync(hbuf, 0, mat * sizeof(float), stream);   // h0 = 0

    dim3 grid(Nn / BN, Nn / BM);                // 8 x 8
    dim3 blk(256);

    for (int t = 0; t < Tt; ++t) {
        // Pz = [x_t|h] @ Wz ; Pr = [x_t|h] @ Wr
        gemm_concat_kernel<<<grid, blk, 0, stream>>>(x, t, hbuf, Wz, Pz);
        gemm_concat_kernel<<<grid, blk, 0, stream>>>(x, t, hbuf, Wr, Pr);
        // z = sigmoid(Pz @ adj + bz) ; rh = sigmoid(Pr @ adj + br) * h
        gemm_adj_kernel<<<grid, blk, 0, stream>>>(Pz, adj, bz, nullptr, hbuf, Zb, 0);
        gemm_adj_kernel<<<grid, blk, 0, stream>>>(Pr, adj, br, nullptr, hbuf, RH, 1);
        // Ph = [x_t|rh] @ Wh
        gemm_concat_kernel<<<grid, blk, 0, stream>>>(x, t, RH, Wh, Ph);
        // h = sigmoid(z*h + (1-z)*tanh(Ph @ adj + bh))   (in place)
        gemm_adj_kernel<<<grid, blk, 0, stream>>>(Ph, adj, bh, Zb, hbuf, nullptr, 2);
        // actions
        actor_kernel<<<dim3(Nn), dim3(64), 0, stream>>>(hbuf, fcw, fcb, out, t);
    }
}